// ConvCapsuleLayer_26972394619639
// MI455X (gfx1250) — compile-verified
//
#include <hip/hip_runtime.h>

#define K5 5
#define CAPS_IN 8
#define CAPS_C 8
#define ROUTINGS 3
#define EPSF 1e-5f
#define HW 48
#define CAPS_I 200           // 5*5*8
#define POS_PER_BLK 4
#define NCH 16

typedef __attribute__((ext_vector_type(2))) float v2f;
typedef __attribute__((ext_vector_type(8))) float v8f;
typedef __attribute__((ext_vector_type(2))) _Float16 h2;
typedef __attribute__((ext_vector_type(4))) _Float16 h4;
typedef __attribute__((ext_vector_type(8))) _Float16 h8;

// NOTE: votes are stored with the n-dimension PERMUTED: n' = (n%4)*4 + n/4
// (4x4 pose transpose, self-inverse). miu/hls/i2s/betav use the same order;
// the final output write applies the inverse permutation.
struct alignas(16) Smem {
    _Float16 votes[POS_PER_BLK][CAPS_I][CAPS_C][NCH]; // 204800 B (first => 16B aligned)
    float r[POS_PER_BLK][CAPS_I][CAPS_C];        // 25600 B
    float act_in[POS_PER_BLK][CAPS_I];           //  3200 B
    float miu[POS_PER_BLK][CAPS_C][NCH];         //  2048 B (permuted n)
    float hls[POS_PER_BLK][CAPS_C][NCH];         //  2048 B  0.5*log(sigma)
    float i2s[POS_PER_BLK][CAPS_C][NCH];         //  2048 B  0.5/sigma
    float act_out[POS_PER_BLK][CAPS_C];          //   128 B
    float rs_part[POS_PER_BLK][64][CAPS_C];      //  8192 B
    float rsum[POS_PER_BLK][CAPS_C];             //   128 B
    float betav[CAPS_C][NCH];                    //   512 B (permuted n)
    float betaa[CAPS_C];                         //    32 B
};

__global__ __launch_bounds__(256, 1)
void capsule_em_kernel(const float* __restrict__ in, const float* __restrict__ Wt,
                       const float* __restrict__ beta_v, const float* __restrict__ beta_a,
                       float* __restrict__ out)
{
    extern __shared__ char smem_raw[];
    Smem& S = *reinterpret_cast<Smem*>(smem_raw);

    const int tid  = threadIdx.x;
    const int wave = tid >> 5;
    const int lane = tid & 31;
    const int b0   = blockIdx.x * POS_PER_BLK;

    // ---------------- stage activations + betas ----------------
    for (int idx = tid; idx < POS_PER_BLK * CAPS_I; idx += 256) {
        int pos = idx / CAPS_I, i = idx % CAPS_I;
        int b = b0 + pos, y = b / HW, x = b % HW;
        int kk = i >> 3, ci = i & 7, ky = kk / K5, kx = kk % K5;
        int sy = y + ky - 2, sx = x + kx - 2;
        int csy = min(max(sy, 0), HW - 1), csx = min(max(sx, 0), HW - 1);
        float msk = (sy == csy && sx == csx) ? 1.f : 0.f;
        S.act_in[pos][i] = msk * in[(csy * HW + csx) * 136 + ci * 17 + 16];
    }
    if (tid < 128) {
        int c = tid >> 4, np = tid & 15;
        int n = ((np & 3) << 2) | (np >> 2);     // inverse (==same) permutation
        S.betav[c][np] = beta_v[c * NCH + n];
    }
    if (tid < 8) S.betaa[tid] = beta_a[tid];

    // ---------------- votes via V_WMMA_F32_16X16X4_F32 ----------------
    // M = (pos4, p) ; N = (c4, r) ; K = q.   Zero-waste f32 WMMA.
    {
        const int mloc = lane & 15;
        const int posA = mloc >> 2, pRow = mloc & 3;
        const int qb   = (lane >> 4) << 1;           // K=0,1 lanes 0-15 ; K=2,3 lanes 16-31
        const int c4   = (lane & 15) >> 2, rr = lane & 3;
        const int bA   = b0 + posA;
        const int yA   = bA / HW, xA = bA % HW;
        const int pos0 = (lane >> 4) << 1;           // D rows: M=Mbase..Mbase+7 -> pos0, pos0+1

        for (int i = wave * 25; i < wave * 25 + 25; ++i) {
            int kk = i >> 3, ci = i & 7, ky = kk / K5, kx = kk % K5;
            int sy = yA + ky - 2, sx = xA + kx - 2;
            int csy = min(max(sy, 0), HW - 1), csx = min(max(sx, 0), HW - 1);
            float msk = (sy == csy && sx == csx) ? 1.f : 0.f;
            const float* src = in + ((csy * HW + csx) * 136 + ci * 17 + (pRow << 2) + qb);
            v2f a; a.x = msk * src[0]; a.y = msk * src[1];
            __builtin_prefetch(src + 136, 0, 0);     // next kx -> global_prefetch_b8
#pragma unroll
            for (int ch = 0; ch < 2; ++ch) {
                int c = (ch << 2) + c4;
                const float* ws = Wt + (((i * CAPS_C + c) * 4 + qb) * 4 + rr);
                v2f bm; bm.x = ws[0]; bm.y = ws[4];
                v8f d = {};
                d = __builtin_amdgcn_wmma_f32_16x16x4_f32(
                        false, a, false, bm, (short)0, d, false, false);
                // D rows M = pos0*4 + p (v=0..3) and (pos0+1)*4 + p (v=4..7).
                // Permuted n' = rr*4 + p  ->  4 contiguous f16 per position.
                h4 lo, hi;
#pragma unroll
                for (int v = 0; v < 4; ++v) { lo[v] = (_Float16)d[v]; hi[v] = (_Float16)d[v + 4]; }
                *reinterpret_cast<h4*>(&S.votes[pos0    ][i][c][rr << 2]) = lo;
                *reinterpret_cast<h4*>(&S.votes[pos0 + 1][i][c][rr << 2]) = hi;
            }
        }
    }
    __syncthreads();

    // ---------------- EM routing: 64 threads per position ----------------
    const int pos = tid >> 6;
    const int t   = tid & 63;

    for (int it = 0; it < ROUTINGS; ++it) {
        // --- Step A: recompute r[i][c] (thread <- i) ---
        float rs_local[CAPS_C];
#pragma unroll
        for (int c = 0; c < CAPS_C; ++c) rs_local[c] = 0.f;

        for (int i = t; i < CAPS_I; i += 64) {
            float rv[CAPS_C];
            if (it == 0) {
#pragma unroll
                for (int c = 0; c < CAPS_C; ++c) rv[c] = 1.f / CAPS_C;
            } else {
                float smax = -1e30f;
                float ssum[CAPS_C];
#pragma unroll
                for (int c = 0; c < CAPS_C; ++c) {
                    const h8* vp = reinterpret_cast<const h8*>(&S.votes[pos][i][c][0]);
                    h8 v0 = vp[0], v1 = vp[1];           // 2x ds_load_b128
                    float s = 0.f;
#pragma unroll
                    for (int n = 0; n < NCH; ++n) {
                        float v   = (n < 8) ? (float)v0[n & 7] : (float)v1[n & 7];
                        float dd  = v - S.miu[pos][c][n];
                        float lp  = -S.hls[pos][c][n] - dd * dd * S.i2s[pos][c][n];
                        s += lp;
                        smax = fmaxf(smax, lp);
                    }
                    ssum[c] = s;
                }
                float shift = 16.f * (smax - 2.302585093f); // 16*(max - ln 10)
                float apsum = 0.f;
#pragma unroll
                for (int c = 0; c < CAPS_C; ++c) {
                    float ap = __expf(ssum[c] - shift) * S.act_out[pos][c];
                    rv[c] = ap; apsum += ap;
                }
                float inv = 1.f / (apsum + EPSF);
#pragma unroll
                for (int c = 0; c < CAPS_C; ++c) rv[c] *= inv;
            }
            float ai = S.act_in[pos][i];
            float rs = 0.f;
#pragma unroll
            for (int c = 0; c < CAPS_C; ++c) { rv[c] *= ai; rs += rv[c]; }
            float inv = 1.f / (rs + EPSF);
#pragma unroll
            for (int c = 0; c < CAPS_C; ++c) {
                rv[c] *= inv;
                S.r[pos][i][c] = rv[c];
                rs_local[c] += rv[c];
            }
        }
#pragma unroll
        for (int c = 0; c < CAPS_C; ++c) S.rs_part[pos][t][c] = rs_local[c];
        __syncthreads();
        if (t < CAPS_C) {
            float s = 0.f;
            for (int j = 0; j < 64; ++j) s += S.rs_part[pos][j][t];
            S.rsum[pos][t] = s;
        }
        __syncthreads();

        // --- Step B: miu / sigma  (thread <- (c, n-pair); one b32 load per i) ---
        {
            int c  = t >> 3;
            int np = t & 7;                 // n' = 2*np, 2*np+1
            float invr = 1.f / (S.rsum[pos][c] + EPSF);
            float m0 = 0.f, m20 = 0.f, m1 = 0.f, m21 = 0.f;
            for (int i = 0; i < CAPS_I; ++i) {
                h2 pv = *reinterpret_cast<const h2*>(&S.votes[pos][i][c][np << 1]);
                float r1 = S.r[pos][i][c] * invr;
                float va = (float)pv.x, vb = (float)pv.y;
                m0  = fmaf(va, r1, m0);
                m20 = fmaf(va * va, r1, m20);
                m1  = fmaf(vb, r1, m1);
                m21 = fmaf(vb * vb, r1, m21);
            }
            float sig0 = fmaxf(m20 - m0 * m0, 0.f) + EPSF;
            float sig1 = fmaxf(m21 - m1 * m1, 0.f) + EPSF;
            int n0 = np << 1;
            S.miu[pos][c][n0]     = m0;
            S.miu[pos][c][n0 + 1] = m1;
            S.hls[pos][c][n0]     = 0.5f * __logf(sig0);
            S.hls[pos][c][n0 + 1] = 0.5f * __logf(sig1);
            S.i2s[pos][c][n0]     = 0.5f / sig0;
            S.i2s[pos][c][n0 + 1] = 0.5f / sig1;
        }
        __syncthreads();

        // --- Step C: activation logits + softmax over c ---
        if (it == ROUTINGS - 1) {
            if (t < CAPS_C) {
                float cost = 0.f;
#pragma unroll
                for (int n = 0; n < NCH; ++n) cost += S.betav[t][n] + S.hls[pos][t][n];
                S.act_out[pos][t] = S.betaa[t] - cost * S.rsum[pos][t];
            }
        } else {
            if (t < CAPS_C) S.act_out[pos][t] = S.rsum[pos][t];
        }
        __syncthreads();
        if (t == 0) {
            float mx = -1e30f;
#pragma unroll
            for (int c = 0; c < CAPS_C; ++c) mx = fmaxf(mx, S.act_out[pos][c]);
            float e[CAPS_C], sm = 0.f;
#pragma unroll
            for (int c = 0; c < CAPS_C; ++c) { e[c] = __expf(S.act_out[pos][c] - mx); sm += e[c]; }
            float inv = 1.f / sm;
#pragma unroll
            for (int c = 0; c < CAPS_C; ++c) S.act_out[pos][c] = e[c] * inv;
        }
        __syncthreads();
    }

    // ---------------- output: [pose(16) | act(1)] per capsule ----------------
    {
        int b = b0 + pos;
        for (int e = t; e < CAPS_C * 17; e += 64) {
            int c = e / 17, k2 = e % 17;
            float val;
            if (k2 == 16) val = S.act_out[pos][c];
            else {
                int np = ((k2 & 3) << 2) | (k2 >> 2);   // inverse n-permutation
                val = S.miu[pos][c][np];
            }
            out[b * 136 + e] = val;
        }
    }
}

extern "C" void kernel_launch(void* const* d_in, const int* in_sizes, int n_in,
                              void* d_out, int out_size, void* d_ws, size_t ws_size,
                              hipStream_t stream) {
    (void)in_sizes; (void)n_in; (void)out_size; (void)d_ws; (void)ws_size;
    const float* in  = (const float*)d_in[0];
    const float* Wt  = (const float*)d_in[1];
    const float* bv  = (const float*)d_in[2];
    const float* ba  = (const float*)d_in[3];
    float* out = (float*)d_out;
    dim3 grid((HW * HW) / POS_PER_BLK);   // 2304/4 = 576 workgroups
    capsule_em_kernel<<<grid, 256, sizeof(Smem), stream>>>(in, Wt, bv, ba, out);
}